// VoxelFeatureExtractor_33749853012245
// MI455X (gfx1250) — compile-verified
//
#include <hip/hip_runtime.h>
#include <stdint.h>

// ---------------- problem constants (match reference) ----------------
#define N_POINTS   200000
#define MAX_PTS    35
#define MAX_VOX    200000
#define GXD        1408
#define GYD        1600
#define GZD        40
#define GYZ        64000u            // GYD*GZD
#define TOTAL_VOX  90112000u         // GXD*GYZ  ( < 2^27 )

#define FEAT_FLOATS (MAX_VOX * MAX_PTS * 4)           // 28,000,000
#define COORD_OFF   FEAT_FLOATS                        // then [MAX_VOX][3]
#define NUM_OFF     (FEAT_FLOATS + MAX_VOX * 3)        // then [MAX_VOX]

// ---------------- radix sort config: 3 passes x 9 bits = 27 bits ------
#define RS_BITS    9
#define RS_DIGITS  512
#define RS_NB      125            // blocks; 125*1600 == 200000 exactly
#define RS_IPB     1600           // items per block
#define RS_T       256            // threads per block (8 wave32's)
#define RS_WAVES   8
#define RS_NBAT    ((RS_IPB + RS_T - 1) / RS_T)        // 7

#define FB_T       256
#define FB_NB      ((N_POINTS + FB_T - 1) / FB_T)      // 782
#define HALO       36
#define TILE       (FB_T + 2 * HALO)                   // 328

static __device__ __forceinline__ unsigned ballot32(bool p) {
    return __builtin_amdgcn_ballot_w32(p);
}

// ---------------- CDNA5 async global -> LDS staging -------------------
#if defined(__has_builtin)
#  if __has_builtin(__builtin_amdgcn_global_load_async_to_lds_b32)
#    define HAVE_ASYNC_LDS 1
#  endif
#endif
#ifndef HAVE_ASYNC_LDS
#  define HAVE_ASYNC_LDS 0
#endif

typedef __attribute__((address_space(1))) int as1_int;
typedef __attribute__((address_space(3))) int as3_int;

static __device__ __forceinline__ void stage_b32(const unsigned* g, unsigned* l) {
#if HAVE_ASYNC_LDS
    // GLOBAL_LOAD_ASYNC_TO_LDS_B32: copies 4 bytes from global address g to LDS address l, tracked with ASYNCcnt
    // builtin prototype: void(int AS1*, int AS3*, Imm int offset, Imm int cpol)
    __builtin_amdgcn_global_load_async_to_lds_b32((as1_int*)g, (as3_int*)l, 0, 0);
#else
    *l = *g;   // synchronous fallback (toolchain without the builtin)
#endif
}
static __device__ __forceinline__ void wait_async_lds() {
    asm volatile("s_wait_asynccnt 0x0" ::: "memory");
}

// ---------------------------------------------------------------------
// 1) zero-fill d_out (dominant HBM cost: 115.2 MB of b128 stores)
// ---------------------------------------------------------------------
__global__ void k_zero(float4* __restrict__ out4, int n4,
                       float* __restrict__ out, int rem_base, int rem) {
    int i = blockIdx.x * blockDim.x + threadIdx.x;
    if (i < n4) out4[i] = make_float4(0.f, 0.f, 0.f, 0.f);
    if (i < rem) out[rem_base + i] = 0.f;
}

// ---------------------------------------------------------------------
// 2) linear voxel keys (sentinel TOTAL_VOX for invalid points)
// ---------------------------------------------------------------------
__global__ void k_keys(const float4* __restrict__ pts,
                       unsigned* __restrict__ keys, unsigned* __restrict__ idx) {
    int i = blockIdx.x * blockDim.x + threadIdx.x;
    if (i >= N_POINTS) return;
    __builtin_prefetch(&pts[i + 256], 0, 0);            // global_prefetch_b8
    float4 p = pts[i];
    // match reference float32 math: floor((p - rmin) / voxel_size)
    float cx = floorf((p.x - 0.0f)  / 0.05f);
    float cy = floorf((p.y + 40.0f) / 0.05f);
    float cz = floorf((p.z + 3.0f)  / 0.1f);
    bool valid = (cx >= 0.f) && (cx < (float)GXD) &&
                 (cy >= 0.f) && (cy < (float)GYD) &&
                 (cz >= 0.f) && (cz < (float)GZD);
    unsigned lin = TOTAL_VOX;
    if (valid)
        lin = (unsigned)cx * GYZ + (unsigned)cy * (unsigned)GZD + (unsigned)cz;
    keys[i] = lin;
    idx[i]  = (unsigned)i;
}

// ---------------------------------------------------------------------
// 3a) per-block 512-bin histogram; double-buffered async LDS staging
// ---------------------------------------------------------------------
__global__ void k_hist(const unsigned* __restrict__ keys,
                       unsigned* __restrict__ hist, int shift) {
    __shared__ unsigned h[RS_DIGITS];
    __shared__ unsigned buf[2][RS_T];
    int t = threadIdx.x;
    for (int d = t; d < RS_DIGITS; d += RS_T) h[d] = 0u;
    int base = blockIdx.x * RS_IPB;
    // prologue: stage batch 0
    {
        int i0 = base + t;
        if (i0 < N_POINTS) stage_b32(keys + i0, &buf[0][t]);
    }
    __syncthreads();
    for (int bt = 0; bt < RS_NBAT; bt++) {
        wait_async_lds();
        __syncthreads();
        int off = bt * RS_T + t;
        bool ok = (off < RS_IPB) && (base + off < N_POINTS);
        unsigned key = buf[bt & 1][t];
        // stage next batch while this one feeds the histogram
        if (bt + 1 < RS_NBAT) {
            int noff = (bt + 1) * RS_T + t;
            int ni = base + noff;
            if (noff < RS_IPB && ni < N_POINTS)
                stage_b32(keys + ni, &buf[(bt + 1) & 1][t]);
        }
        if (ok) atomicAdd(&h[(key >> shift) & (RS_DIGITS - 1)], 1u);
        __syncthreads();
    }
    for (int d = t; d < RS_DIGITS; d += RS_T)
        hist[d * RS_NB + blockIdx.x] = h[d];
}

// ---------------------------------------------------------------------
// 3b) exclusive scan of the 64000-entry histogram (single block, 2-level)
// ---------------------------------------------------------------------
__global__ void k_scan_hist(unsigned* __restrict__ hist) {
    __shared__ unsigned sums[RS_T];
    const int M = RS_DIGITS * RS_NB;   // 64000
    const int C = M / RS_T;            // 250
    int t = threadIdx.x;
    unsigned s = 0;
    for (int e = 0; e < C; e++) s += hist[t * C + e];
    sums[t] = s;
    __syncthreads();
    if (t == 0) {
        unsigned run = 0;
        for (int k = 0; k < RS_T; k++) { unsigned v = sums[k]; sums[k] = run; run += v; }
    }
    __syncthreads();
    unsigned run = sums[t];
    for (int e = 0; e < C; e++) {
        unsigned v = hist[t * C + e];
        hist[t * C + e] = run;
        run += v;
    }
}

// ---------------------------------------------------------------------
// 3c) stable scatter: wave32 ballot peer-matching for in-order ranks
// ---------------------------------------------------------------------
__global__ void k_scatter(const unsigned* __restrict__ keysIn,
                          const unsigned* __restrict__ idxIn,
                          unsigned* __restrict__ keysOut,
                          unsigned* __restrict__ idxOut,
                          const unsigned* __restrict__ hist, int shift) {
    __shared__ unsigned s_gbase[RS_DIGITS];
    __shared__ unsigned s_dbase[RS_DIGITS];
    __shared__ unsigned s_total[RS_DIGITS];
    __shared__ unsigned s_wcnt[RS_DIGITS * RS_WAVES];   // 16 KB
    int t = threadIdx.x, b = blockIdx.x;
    for (int d = t; d < RS_DIGITS; d += RS_T) {
        s_gbase[d] = hist[d * RS_NB + b];
        s_dbase[d] = 0u;
    }
    int base = b * RS_IPB;
    unsigned lane = t & 31u;
    int wave = t >> 5;
    unsigned below = (1u << lane) - 1u;
    for (int bt = 0; bt < RS_NBAT; bt++) {
        int off = bt * RS_T + t;
        int i = base + off;
        bool act = (off < RS_IPB) && (i < N_POINTS);
        unsigned key = 0u, pay = 0u, dig = 0u;
        if (act) {
            key = keysIn[i];
            pay = idxIn[i];
            dig = (key >> shift) & (RS_DIGITS - 1);
        }
        for (int z = t; z < RS_DIGITS * RS_WAVES; z += RS_T) s_wcnt[z] = 0u;
        __syncthreads();
        // lanes sharing my 9-bit digit (wave32 mask)
        unsigned peers = ballot32(act);
        #pragma unroll
        for (int bit = 0; bit < RS_BITS; bit++) {
            unsigned bal = ballot32(((dig >> bit) & 1u) != 0u);
            peers &= ((dig >> bit) & 1u) ? bal : ~bal;
        }
        unsigned lrank = __popc(peers & below);
        if (act && lrank == 0u)
            s_wcnt[dig * RS_WAVES + wave] = (unsigned)__popc(peers);
        __syncthreads();
        // per-digit exclusive scan across the 8 waves
        for (int d = t; d < RS_DIGITS; d += RS_T) {
            unsigned run = 0;
            for (int w = 0; w < RS_WAVES; w++) {
                unsigned v = s_wcnt[d * RS_WAVES + w];
                s_wcnt[d * RS_WAVES + w] = run;
                run += v;
            }
            s_total[d] = run;
        }
        __syncthreads();
        if (act) {
            unsigned pos = s_gbase[dig] + s_dbase[dig] +
                           s_wcnt[dig * RS_WAVES + wave] + lrank;
            keysOut[pos] = key;
            idxOut[pos]  = pay;
        }
        __syncthreads();
        for (int d = t; d < RS_DIGITS; d += RS_T) s_dbase[d] += s_total[d];
        __syncthreads();
    }
}

// ---------------------------------------------------------------------
// 4a) per-block count of "first point of a valid voxel" flags
// ---------------------------------------------------------------------
__global__ void k_flagsums(const unsigned* __restrict__ keys,
                           unsigned* __restrict__ bsums) {
    __shared__ unsigned s_w[FB_T / 32];
    int i = blockIdx.x * FB_T + threadIdx.x;
    bool f = false;
    if (i < N_POINTS) {
        unsigned k = keys[i];
        f = (k < TOTAL_VOX) && (i == 0 || keys[i - 1] != k);
    }
    unsigned m = ballot32(f);
    unsigned lane = threadIdx.x & 31u;
    int wave = threadIdx.x >> 5;
    if (lane == 0) s_w[wave] = (unsigned)__popc(m);
    __syncthreads();
    if (threadIdx.x == 0) {
        unsigned s = 0;
        for (int w = 0; w < FB_T / 32; w++) s += s_w[w];
        bsums[blockIdx.x] = s;
    }
}

// 4b) exclusive scan of 782 block sums (tiny; serial is fine)
__global__ void k_scan_bsums(unsigned* __restrict__ bsums, int nb) {
    if (threadIdx.x == 0 && blockIdx.x == 0) {
        unsigned run = 0;
        for (int k = 0; k < nb; k++) { unsigned v = bsums[k]; bsums[k] = run; run += v; }
    }
}

// ---------------------------------------------------------------------
// 5) finalize: async-stage sorted keys (+halo, sentinel-padded) into LDS,
//    then all neighbor scans (rank / count / segment-head test) hit LDS.
// ---------------------------------------------------------------------
__global__ void k_finalize(const unsigned* __restrict__ keys,
                           const unsigned* __restrict__ idx,
                           const float4* __restrict__ pts,
                           const unsigned* __restrict__ boff,
                           float* __restrict__ out) {
    __shared__ unsigned s_w[FB_T / 32];
    __shared__ unsigned s_key[TILE];
    int t = threadIdx.x;
    int i = blockIdx.x * FB_T + t;
    int tile_base = blockIdx.x * FB_T - HALO;
    // stage tile: in-range via async global->LDS, out-of-range = sentinel
    for (int j = t; j < TILE; j += FB_T) {
        int g = tile_base + j;
        if (g >= 0 && g < N_POINTS) stage_b32(keys + g, &s_key[j]);
        else                        s_key[j] = TOTAL_VOX;
    }
    wait_async_lds();
    __syncthreads();

    unsigned key = s_key[HALO + t];           // == keys[i] (sentinel if i>=N)
    bool valid = key < TOTAL_VOX;
    bool f = valid && (s_key[HALO + t - 1] != key);   // halo sentinel => i==0 ok
    unsigned m = ballot32(f);
    unsigned lane = t & 31u;
    int wave = t >> 5;
    if (lane == 0) s_w[wave] = (unsigned)__popc(m);
    __syncthreads();
    unsigned wex = 0;
    for (int w = 0; w < wave; w++) wex += s_w[w];
    // E = # of segment-heads strictly before position i (global)
    unsigned E = boff[blockIdx.x] + wex + (unsigned)__popc(m & ((1u << lane) - 1u));
    if (!valid) return;
    unsigned slot = f ? E : (E - 1u);
    if (slot >= MAX_VOX) return;
    // within-voxel rank: consecutive predecessors with same key (LDS probes;
    // sentinel halo terminates the loop, no bounds checks needed)
    int r = 0;
    while (r < MAX_PTS && s_key[HALO + t - 1 - r] == key) r++;
    if (r < MAX_PTS) {
        float4 p = pts[idx[i]];
        ((float4*)out)[(size_t)slot * MAX_PTS + r] = p;   // 16B-aligned store
    }
    if (f) {
        int c = 1;
        while (c < MAX_PTS && s_key[HALO + t + c] == key) c++;
        out[NUM_OFF + slot] = (float)c;                    // min(count, 35)
        unsigned x = key / GYZ, rem = key % GYZ;
        out[COORD_OFF + (size_t)slot * 3 + 0] = (float)x;
        out[COORD_OFF + (size_t)slot * 3 + 1] = (float)(rem / (unsigned)GZD);
        out[COORD_OFF + (size_t)slot * 3 + 2] = (float)(rem % (unsigned)GZD);
    }
}

// ---------------------------------------------------------------------
extern "C" void kernel_launch(void* const* d_in, const int* in_sizes, int n_in,
                              void* d_out, int out_size, void* d_ws, size_t ws_size,
                              hipStream_t stream) {
    const float4* pts = (const float4*)d_in[0];
    float* out = (float*)d_out;

    // workspace carve-out (~3.5 MB)
    unsigned* keysA = (unsigned*)d_ws;
    unsigned* keysB = keysA + N_POINTS;
    unsigned* idxA  = keysB + N_POINTS;
    unsigned* idxB  = idxA + N_POINTS;
    unsigned* hist  = idxB + N_POINTS;                 // RS_DIGITS*RS_NB = 64000
    unsigned* bsums = hist + RS_DIGITS * RS_NB;        // FB_NB = 782

    // 1) zero the full output (padding semantics)
    int n4 = out_size / 4, rem = out_size - n4 * 4;
    int zb = (n4 + 255) / 256;
    if (zb < 1) zb = 1;
    k_zero<<<zb, 256, 0, stream>>>((float4*)out, n4, out, n4 * 4, rem);

    // 2) keys
    k_keys<<<(N_POINTS + 255) / 256, 256, 0, stream>>>(pts, keysA, idxA);

    // 3) stable LSD radix sort: 3 x 9-bit passes over 27-bit keys
    unsigned *ki = keysA, *pi = idxA, *ko = keysB, *po = idxB;
    for (int pass = 0; pass < 3; pass++) {
        int shift = pass * RS_BITS;
        k_hist<<<RS_NB, RS_T, 0, stream>>>(ki, hist, shift);
        k_scan_hist<<<1, RS_T, 0, stream>>>(hist);
        k_scatter<<<RS_NB, RS_T, 0, stream>>>(ki, pi, ko, po, hist, shift);
        unsigned* tk = ki; ki = ko; ko = tk;
        unsigned* tp = pi; pi = po; po = tp;
    }
    // sorted (key,origIdx) now in (ki, pi)

    // 4) slot prefix over is_first flags
    k_flagsums<<<FB_NB, FB_T, 0, stream>>>(ki, bsums);
    k_scan_bsums<<<1, 64, 0, stream>>>(bsums, FB_NB);

    // 5) scatter voxel features / coords / counts
    k_finalize<<<FB_NB, FB_T, 0, stream>>>(ki, pi, pts, bsums, out);
}